// IntraConvUpsample_8375186227642
// MI455X (gfx1250) — compile-verified
//
#include <hip/hip_runtime.h>
#include <math.h>

typedef __attribute__((ext_vector_type(2))) float v2f;
typedef __attribute__((ext_vector_type(4))) float v4f;
typedef __attribute__((ext_vector_type(8))) float v8f;

#define Bb 2
#define Cc 16
#define Dd 24
#define Ss 3
#define Oo 16
#define Kk 3
#define Tt 48
#define XO 22                      // D - K + 1
#define VOX (XO*XO*XO)             // 10648
#define KC  (Kk*Kk*Kk*Cc)          // 432
#define EPSF 1e-6f

// strides of x: [B, C, D, D, D, S]
#define XSB (Cc*Dd*Dd*Dd*Ss)       // 663552
#define XSC (Dd*Dd*Dd*Ss)          // 41472
#define XSX (Dd*Dd*Ss)             // 1728
#define XSY (Dd*Ss)                // 72

// -------- Kernel 1: channel sum, vectorized b128: xs[b,t..t+3] = sum_c x[b,c,t..t+3] --------
__global__ void k_chansum(const float* __restrict__ x, float* __restrict__ xs) {
    int i = blockIdx.x * 256 + threadIdx.x;              // one float4 per thread
    if (i >= (Bb * XSC) / 4) return;                     // 20736 quads
    int b = i / (XSC / 4);
    int t = (i - b * (XSC / 4)) * 4;
    const float* p = x + (long long)b * XSB + t;
    v4f s = {};
#pragma unroll
    for (int c = 0; c < Cc; ++c) {
        v4f q = *(const v4f*)(p + c * XSC);              // global_load_b128
        s.x += q.x; s.y += q.y; s.z += q.z; s.w += q.w;
    }
    *(v4f*)(xs + b * XSC + t) = s;
}

// -------- Kernel 2: mu = normalize(mean over 3x3x3 window of xs / (27*16)) --------
__global__ void k_mu(const float* __restrict__ xs, float* __restrict__ mu) {
    int v = blockIdx.x * 256 + threadIdx.x;
    if (v >= Bb * VOX) return;                           // 21296 voxels
    int b = v / VOX;
    int r = v - b * VOX;
    int X = r / (XO * XO), Y = (r / XO) % XO, Z = r % XO;
    const float* p = xs + b * XSC + X * XSX + Y * XSY + Z * Ss;
    float m0 = 0.f, m1 = 0.f, m2 = 0.f;
#pragma unroll
    for (int i = 0; i < Kk; ++i)
#pragma unroll
        for (int j = 0; j < Kk; ++j)
#pragma unroll
            for (int l = 0; l < Kk; ++l) {
                const float* q = p + i * XSX + j * XSY + l * Ss;
                m0 += q[0]; m1 += q[1]; m2 += q[2];
            }
    const float inv = 1.f / (float)(Kk * Kk * Kk * Cc);
    m0 *= inv; m1 *= inv; m2 *= inv;
    float n = sqrtf(m0 * m0 + m1 * m1 + m2 * m2);
    float d = fmaxf(n, EPSF);
    mu[v * 3 + 0] = m0 / d;
    mu[v * 3 + 1] = m1 / d;
    mu[v * 3 + 2] = m2 / d;
}

// -------- Kernel 3: fused log-map + WMMA GEMM + exp-map --------
// GEMM: D[16 x 16cols] += W[16 x 432] * logv[432 x 16cols], tiled K=4 per WMMA.
// Columns = (voxel, s) padded to 4 slots/voxel -> exactly 4 voxels per tile.
// Loop order k(27 taps, rolled) x c4(4 channel-blocks, unrolled) -> 4 static WMMAs.
__global__ __launch_bounds__(256) void k_conv_wmma(const float* __restrict__ x,
                                                   const float* __restrict__ w,
                                                   const float* __restrict__ mu,
                                                   float* __restrict__ pre) {
    __shared__ float lw[Oo * KC];                        // 27648 B of 320 KB LDS
    for (int t = threadIdx.x; t < Oo * KC; t += 256) lw[t] = w[t];
    __syncthreads();

    const int lane = threadIdx.x & 31;
    const int wave = threadIdx.x >> 5;
    const int NT = (Bb * VOX * 4) / 16;                  // 5324 tiles
    int tile = blockIdx.x * 8 + wave;
    if (tile >= NT) tile = NT - 1;                       // duplicate work, same result

    const int col = tile * 16 + (lane & 15);
    const int vg  = col >> 2;                            // global voxel id
    const int s   = col & 3;                             // 0..2 real, 3 = zero pad
    const int b   = vg / VOX;
    const int r   = vg - b * VOX;
    const int X = r / (XO * XO), Y = (r / XO) % XO, Z = r % XO;

    const float m0 = mu[vg * 3 + 0];
    const float m1 = mu[vg * 3 + 1];
    const float m2 = mu[vg * 3 + 2];
    const float ms = (s == 0) ? m0 : ((s == 1) ? m1 : ((s == 2) ? m2 : 0.f));

    const float* xb = x + (long long)b * XSB + X * XSX + Y * XSY + Z * Ss;
    const int rofs = (lane >> 4) << 1;                   // lanes 0-15: K rows {0,1}; 16-31: {2,3}
    const int lw_row = (lane & 15) * KC + rofs;          // A-fragment row base in LDS

    v8f acc = {};
    for (int k = 0; k < Kk * Kk * Kk; ++k) {             // 27 window taps
        const int i = k / 9, j = (k / 3) % 3, l = k % 3;
        const float* qk = xb + i * XSX + j * XSY + l * Ss;   // hoisted tap base
        const int lwk = lw_row + k * 16;
#pragma unroll
        for (int c4 = 0; c4 < Cc; c4 += 4) {             // 4 WMMAs per tap
            v2f bfrag;
#pragma unroll
            for (int u = 0; u < 2; ++u) {
                const int c = c4 + rofs + u;
                const float* q = qk + c * XSC;
                const float w0 = q[0], w1 = q[1], w2 = q[2];   // contiguous b96
                float dot = fminf(fmaxf(w0 * m0 + w1 * m1 + w2 * m2,
                                        -1.f + EPSF), 1.f - EPSF);
                float th  = acosf(dot);                  // th in (1.4e-3, pi-1.4e-3)
                float sc  = th / __sinf(th);             // hw v_sin, bounded arg
                float wsv = (s == 0) ? w0 : ((s == 1) ? w1 : ((s == 2) ? w2 : 0.f));
                float lv  = (wsv - dot * ms) * sc;       // s==3 pad -> exactly 0
                if (u == 0) bfrag.x = lv; else bfrag.y = lv;
            }
            const v2f afrag = *(const v2f*)&lw[lwk + c4];    // 8B-aligned ds_load_b64
            acc = __builtin_amdgcn_wmma_f32_16x16x4_f32(false, afrag, false, bfrag,
                                                        (short)0, acc, false, false);
        }
    }

    // Epilogue: rho over the aligned 4-lane voxel group (pad column is exact 0),
    // exp map, scatter to pre-upsample buffer [B, O, Vox, S].
    const int obase = (lane < 16) ? 0 : 8;
#pragma unroll
    for (int e = 0; e < 8; ++e) {
        float vv = acc[e];
        float sq = vv * vv;
        sq += __shfl_xor(sq, 1, 32);
        sq += __shfl_xor(sq, 2, 32);
        float rho = sqrtf(sq);
        float snc = (rho < EPSF) ? 1.f : sinf(rho) / fmaxf(rho, EPSF);
        float outv = cosf(rho) * ms + snc * vv;
        if (s < 3) {
            int o = obase + e;
            pre[(((b * Oo + o) * VOX) + r) * 3 + s] = outv;
        }
    }
}

// -------- Kernel 4: trilinear upsample 22 -> 48, align_corners=True --------
__global__ void k_upsample(const float* __restrict__ pre, float* __restrict__ out) {
    int i = blockIdx.x * 256 + threadIdx.x;
    const int N = Bb * Oo * Tt * Tt * Tt * Ss;           // 10,616,832
    if (i >= N) return;
    int s = i % 3; int q = i / 3;
    int tz = q % Tt; q /= Tt;
    int ty = q % Tt; q /= Tt;
    int tx = q % Tt; q /= Tt;
    int o = q % Oo;  int b = q / Oo;

    const float sc = (float)(XO - 1) / (float)(Tt - 1);
    float px = tx * sc, py = ty * sc, pz = tz * sc;
    int x0 = (int)px, y0 = (int)py, z0 = (int)pz;
    int x1 = min(x0 + 1, XO - 1), y1 = min(y0 + 1, XO - 1), z1 = min(z0 + 1, XO - 1);
    float fx = px - x0, fy = py - y0, fz = pz - z0;

    const float* base = pre + ((b * Oo + o) * VOX) * 3 + s;
#define AT(xi, yi, zi) base[(((xi) * XO + (yi)) * XO + (zi)) * 3]
    float c000 = AT(x0, y0, z0), c001 = AT(x0, y0, z1);
    float c010 = AT(x0, y1, z0), c011 = AT(x0, y1, z1);
    float c100 = AT(x1, y0, z0), c101 = AT(x1, y0, z1);
    float c110 = AT(x1, y1, z0), c111 = AT(x1, y1, z1);
#undef AT
    float c00 = c000 * (1.f - fz) + c001 * fz;
    float c01 = c010 * (1.f - fz) + c011 * fz;
    float c10 = c100 * (1.f - fz) + c101 * fz;
    float c11 = c110 * (1.f - fz) + c111 * fz;
    float c0 = c00 * (1.f - fy) + c01 * fy;
    float c1 = c10 * (1.f - fy) + c11 * fy;
    out[i] = c0 * (1.f - fx) + c1 * fx;
}

extern "C" void kernel_launch(void* const* d_in, const int* in_sizes, int n_in,
                              void* d_out, int out_size, void* d_ws, size_t ws_size,
                              hipStream_t stream) {
    const float* x = (const float*)d_in[0];     // [2,16,24,24,24,3]
    const float* w = (const float*)d_in[1];     // [16, 432]
    float* ws  = (float*)d_ws;
    float* xs  = ws;                            // 82,944 floats
    float* mu  = ws + Bb * XSC;                 // 63,888 floats
    float* pre = mu + Bb * VOX * 3;             // 1,022,208 floats
    float* out = (float*)d_out;

    k_chansum<<<((Bb * XSC) / 4 + 255) / 256, 256, 0, stream>>>(x, xs);
    k_mu<<<(Bb * VOX + 255) / 256, 256, 0, stream>>>(xs, mu);
    const int NT = (Bb * VOX * 4) / 16;         // 5324 WMMA tiles
    k_conv_wmma<<<(NT + 7) / 8, 256, 0, stream>>>(x, w, mu, pre);
    const int N = Bb * Oo * Tt * Tt * Tt * Ss;
    k_upsample<<<(N + 255) / 256, 256, 0, stream>>>(pre, out);
}